// SigMMD_naive_63823214018692
// MI455X (gfx1250) — compile-verified
//
#include <hip/hip_runtime.h>
#include <hip/hip_bf16.h>

// SigMMD fused kernel for MI455X (gfx1250, wave32).
// One workgroup per (gram, a, b) pair: WMMA f32 16x16x4 computes the 64x64
// inc tile (dX . dY^T) in LDS, then wave 0 runs the Goursat PDE recursion
// as 63 wave32 prefix-scans and atomically accumulates the weighted corner.

typedef __attribute__((ext_vector_type(2))) float v2f;
typedef __attribute__((ext_vector_type(8))) float v8f;

#define NPATH 48
#define LLEN  64
#define DCH   64
#define M1    63            // L-1
#define PAIRS (NPATH * NPATH)   // 2304
#define LDSS  66            // padded LDS row stride (dwords)

// ---- kernel 0: base term = mean((X[:,0,:]-Y[:,0,:])^2), overwrites d_out[0]
__global__ void sig_init_out(const float* __restrict__ X,
                             const float* __restrict__ Y,
                             float* __restrict__ out) {
    __shared__ float red[256];
    float s = 0.0f;
    for (int idx = threadIdx.x; idx < NPATH * DCH; idx += 256) {
        int a = idx >> 6, d = idx & 63;
        float diff = X[a * LLEN * DCH + d] - Y[a * LLEN * DCH + d];
        s += diff * diff;
    }
    red[threadIdx.x] = s;
    __syncthreads();
    for (int st = 128; st > 0; st >>= 1) {
        if (threadIdx.x < st) red[threadIdx.x] += red[threadIdx.x + st];
        __syncthreads();
    }
    if (threadIdx.x == 0) out[0] = red[0] / (float)(NPATH * DCH);
}

// ---- kernel 1: fused inc-GEMM (WMMA) + signature PDE recursion
__global__ void __launch_bounds__(128)
sig_pair_kernel(const float* __restrict__ X,
                const float* __restrict__ Y,
                float* __restrict__ out) {
    __shared__ float dXs[64 * LDSS];   // dX tile (row 63 zero-padded)
    __shared__ float dYs[64 * LDSS];   // dY tile (row 63 zero-padded)
    __shared__ float incs[64 * LDSS];  // inc = dX . dY^T

    const int bid = blockIdx.x;
    const int g   = bid / PAIRS;        // 0: XX, 1: YY, 2: XY
    const int p   = bid % PAIRS;
    const int a   = p / NPATH;
    const int b   = p % NPATH;

    const float* Pa = (g == 1) ? Y : X;
    const float* Pb = (g == 0) ? X : Y;
    const float* Ap = Pa + a * LLEN * DCH;
    const float* Bp = Pb + b * LLEN * DCH;
    const float weight = (g == 2) ? (-2.0f / (float)PAIRS)
                                  : ( 1.0f / (float)PAIRS);

    // --- load + diff into LDS (X/Y are L2-resident; 32KB per block) ---
    for (int idx = threadIdx.x; idx < M1 * DCH; idx += 128) {
        int i = idx >> 6, d = idx & 63;
        dXs[i * LDSS + d] = Ap[(i + 1) * DCH + d] - Ap[i * DCH + d];
        dYs[i * LDSS + d] = Bp[(i + 1) * DCH + d] - Bp[i * DCH + d];
    }
    for (int idx = threadIdx.x; idx < 64; idx += 128) {
        dXs[M1 * LDSS + idx] = 0.0f;
        dYs[M1 * LDSS + idx] = 0.0f;
    }
    __syncthreads();

    // --- WMMA phase: wave w computes 16-row band w of the 64x64 inc tile ---
    const int wv   = threadIdx.x >> 5;  // wave id = row-band
    const int lane = threadIdx.x & 31;
    const int m    = lane & 15;         // row (A) / col (B) within tile
    const int kh   = lane >> 4;         // K-half select per ISA layout

    for (int ct = 0; ct < 4; ++ct) {
        v8f acc = {};
#pragma unroll
        for (int k0 = 0; k0 < DCH; k0 += 4) {
            const int kk = k0 + 2 * kh;
            v2f afrag = *(const v2f*)&dXs[(wv * 16 + m) * LDSS + kk];
            v2f bfrag = *(const v2f*)&dYs[(ct * 16 + m) * LDSS + kk];
            acc = __builtin_amdgcn_wmma_f32_16x16x4_f32(
                false, afrag, false, bfrag, (short)0, acc, false, false);
        }
#pragma unroll
        for (int v = 0; v < 8; ++v) {
            int row = wv * 16 + v + 8 * kh;
            int col = ct * 16 + (lane & 15);
            incs[row * LDSS + col] = acc[v];
        }
    }
    __syncthreads();

    // --- PDE recursion on wave 0: K[i+1,j+1] = K[i+1,j] + K[i,j+1]
    //     + K[i,j]*(inc[i,j]-1). Each row is an inclusive prefix sum of
    //     c[j] = K[i,j+1] + K[i,j]*(inc[i,j]-1); lane l owns c[2l], c[2l+1].
    if (threadIdx.x < 32) {
        float r_odd  = 1.0f;  // K[i][2l+1]
        float r_even = 1.0f;  // K[i][2l+2]
        for (int i = 0; i < M1; ++i) {
            float p0 = __shfl_up(r_even, 1);       // K[i][2l] from lane l-1
            if (lane == 0) p0 = 1.0f;              // K[i][0] = 1
            const float p1 = r_odd;
            const float p2 = r_even;
            v2f iv = *(const v2f*)&incs[i * LDSS + 2 * lane];
            float c0 = p1 + p0 * (iv.x - 1.0f);
            float c1 = (lane < 31) ? (p2 + p1 * (iv.y - 1.0f)) : 0.0f;
            // inclusive wave32 scan of pair sums
            float s = c0 + c1;
#pragma unroll
            for (int off = 1; off < 32; off <<= 1) {
                float t = __shfl_up(s, off);
                if (lane >= off) s += t;
            }
            r_odd  = 1.0f + s - c1;   // K[i+1][2l+1]
            r_even = 1.0f + s;        // K[i+1][2l+2]
        }
        // K[63][63] lives in lane 31's r_odd
        if (lane == 31) atomicAdd(out, weight * r_odd);
    }
}

extern "C" void kernel_launch(void* const* d_in, const int* in_sizes, int n_in,
                              void* d_out, int out_size, void* d_ws, size_t ws_size,
                              hipStream_t stream) {
    (void)in_sizes; (void)n_in; (void)out_size; (void)d_ws; (void)ws_size;
    const float* X = (const float*)d_in[0];
    const float* Y = (const float*)d_in[1];
    float* out = (float*)d_out;

    sig_init_out<<<1, 256, 0, stream>>>(X, Y, out);
    sig_pair_kernel<<<3 * PAIRS, 128, 0, stream>>>(X, Y, out);
}